// TemporalPooling_79989470921114
// MI455X (gfx1250) — compile-verified
//
#include <hip/hip_runtime.h>

typedef __attribute__((ext_vector_type(2))) float v2f;
typedef __attribute__((ext_vector_type(8))) float v8f;

#define NB 128   // batches
#define NC 400   // channels
#define NL 100   // length
#define NP 16    // output positions
#define KP 104   // padded K row stride for W rows in LDS (8B-aligned float2s)
#define NKB (NL / 4)    // 25 K-steps of 4 (exact)
#define NTILE (NC / 16) // 25 M-tiles of 16 (exact)

// ---------------------------------------------------------------------------
// Fused kernel, one block (8 wave32) per batch b:
//   Phase 1: build weight matrix W[b] (NP x NL) in LDS (interp weights,
//            j<d mask, 1/max(d,1) folded in). Lane l (<16) owns row l.
//   Phase 2: out[b] (400x16) = x[b] (400x100) @ W[b]^T via
//            V_WMMA_F32_16X16X4_F32.
//            - B fragments (25 x float2 = 50 VGPRs) preloaded once per wave.
//            - A fragments of each tile staged into 25 distinct float2 regs
//              so all 25 global_load_b64 are in flight before the WMMA chain
//              (memory-level parallelism; avoids load->wait0->wmma lockstep).
//            - Two accumulator chains (even/odd K) to pipeline WMMA latency.
// WMMA f32 16x16x4 layouts (wave32):
//   A: lane = M + 16*(K>=2), vgpr = K&1   -> float2 global load from x
//   B: lane = N + 16*(K>=2), vgpr = K&1   -> float2 from LDS-built W[b]
//   D: vgpr r -> M = r + 8*(lane>=16), N = lane&15
// ---------------------------------------------------------------------------
__global__ __launch_bounds__(256) void tp_fused_wmma_kernel(
    const float* __restrict__ x, const int* __restrict__ seg,
    float* __restrict__ out) {
  __shared__ float wlds[NP * KP];
  const int b   = blockIdx.x;
  const int tid = threadIdx.x;

  // ---- Phase 1: build W[b] in LDS -------------------------------------
  for (int i = tid; i < NP * KP; i += 256) wlds[i] = 0.0f;
  __syncthreads();

  if (tid < NP) {
    const int s = seg[2 * b + 0];
    const int e = seg[2 * b + 1];
    const int d = e - s;
    const float inv  = 1.0f / (float)((d > 1) ? d : 1);
    const int   l    = tid;
    const int   jmax = (d < NL) ? d : NL;        // mask: j < d
    const int   base = NP * s + NP / 2 + d * l;  // pos = 16*s + 8 + d*l + j
    const float xmin = 0.5f * (float)NP;                 // 8.0
    const float xmax = ((float)NL - 0.5f) * (float)NP;   // 1592.0
    for (int j = 0; j < jmax; ++j) {
      float pos = (float)(base + j);
      float t   = fminf(fmaxf(pos, xmin), xmax) * (1.0f / (float)NP) - 0.5f;
      int   i0  = (int)floorf(t);
      i0        = (i0 < 0) ? 0 : ((i0 > NL - 2) ? NL - 2 : i0);
      float fr  = t - (float)i0;
      wlds[l * KP + i0]     += (1.0f - fr) * inv;
      wlds[l * KP + i0 + 1] += fr * inv;
    }
  }
  __syncthreads();

  // ---- Phase 2: WMMA GEMM ---------------------------------------------
  const int lane = tid & 31;
  const int wave = __builtin_amdgcn_readfirstlane(tid >> 5);  // SGPR-uniform
  const int m    = lane & 15;  // M (A) / N (B, D) index
  const int khi  = lane >> 4;  // K half: 0 -> K{0,1}, 1 -> K{2,3}

  // Preload all B fragments for this wave's lane role (reused by every tile).
  v2f bfrag[NKB];
  {
    const float* brow = &wlds[m * KP + 2 * khi];
#pragma unroll
    for (int kb = 0; kb < NKB; ++kb)
      bfrag[kb] = *(const v2f*)(brow + kb * 4);  // ds_load_b64 (8B aligned)
  }

  const float* xb = x + (size_t)b * NC * NL;
  float*       ob = out + (size_t)b * NC * NP;

  for (int tile = wave; tile < NTILE; tile += 8) {  // scalar loop, EXEC=all-1
    const float* arow = xb + (size_t)(tile * 16 + m) * NL + 2 * khi;

    // Stage all A fragments first: 25 independent b64 loads in flight.
    v2f a[NKB];
#pragma unroll
    for (int kb = 0; kb < NKB; ++kb)
      a[kb] = *(const v2f*)(arow + kb * 4);  // global_load_b64 (8B aligned)

    // Two interleaved accumulation chains over K.
    v8f c0 = {};
    v8f c1 = {};
#pragma unroll
    for (int kb = 0; kb < NKB; kb += 2) {
      c0 = __builtin_amdgcn_wmma_f32_16x16x4_f32(
          false, a[kb], false, bfrag[kb], (short)0, c0, false, false);
      if (kb + 1 < NKB)
        c1 = __builtin_amdgcn_wmma_f32_16x16x4_f32(
            false, a[kb + 1], false, bfrag[kb + 1], (short)0, c1, false, false);
    }
    v8f c = c0 + c1;

    const int row0 = tile * 16 + 8 * khi;  // vgpr r -> row row0 + r, col m
#pragma unroll
    for (int r = 0; r < 8; ++r) {
      ob[(size_t)(row0 + r) * NP + m] = c[r];
    }
  }
}

// ---------------------------------------------------------------------------
extern "C" void kernel_launch(void* const* d_in, const int* in_sizes, int n_in,
                              void* d_out, int out_size, void* d_ws,
                              size_t ws_size, hipStream_t stream) {
  const float* x   = (const float*)d_in[0];  // (128, 400, 100) f32
  const int*   seg = (const int*)d_in[1];    // (128, 2) i32
  float*       out = (float*)d_out;          // (128, 400, 16) f32

  (void)in_sizes; (void)n_in; (void)out_size; (void)d_ws; (void)ws_size;

  tp_fused_wmma_kernel<<<NB, 256, 0, stream>>>(x, seg, out);
}